// MultiHeadAttention_41601053229684
// MI455X (gfx1250) — compile-verified
//
#include <hip/hip_runtime.h>
#include <hip/hip_bf16.h>

typedef __attribute__((ext_vector_type(2))) float v2f;
typedef __attribute__((ext_vector_type(8))) float v8f;

// Problem constants (fixed by the reference setup)
constexpr int DM = 1024;   // d_model
constexpr int SEQ = 2048;  // sequence length
constexpr int BATCH = 2;
constexpr int H = 16;      // heads
constexpr int DK = 64;     // head dim
constexpr int ROWS = BATCH * SEQ;        // 4096 tokens
constexpr int MT = ROWS / 16;            // 256 row tiles
constexpr int NT = DM / 16;              // 64  col tiles

static __device__ __forceinline__ v8f wmma_f32(v2f a, v2f b, v8f c) {
  // V_WMMA_F32_16X16X4_F32 : D(16x16 f32) = A(16x4 f32) * B(4x16 f32) + C
  return __builtin_amdgcn_wmma_f32_16x16x4_f32(
      /*neg_a=*/false, a, /*neg_b=*/false, b,
      /*c_mod=*/(short)0, c, /*reuse_a=*/false, /*reuse_b=*/false);
}

// ---------------------------------------------------------------------------
// Kernel 1: fused QKV projection.  y = x @ W.T + b, reshaped to [B,H,S,DK].
// One wave computes one 16x16 output tile, marching K=1024 in steps of 4
// with the native fp32 WMMA.  A: lane = M(row), VGPR pair = K.  B: lane =
// N(col of W row), VGPR pair = K.  Both are contiguous-K -> single b64 loads.
// ---------------------------------------------------------------------------
__global__ __launch_bounds__(256) void proj_kernel(
    const float* __restrict__ Xq, const float* __restrict__ Xk,
    const float* __restrict__ Xv,
    const float* __restrict__ Wq, const float* __restrict__ bq,
    const float* __restrict__ Wk, const float* __restrict__ bk,
    const float* __restrict__ Wv, const float* __restrict__ bv,
    float* __restrict__ Qh, float* __restrict__ Kh, float* __restrict__ Vh) {
  const int lane = threadIdx.x & 31;
  const int wave = threadIdx.x >> 5;
  const int t = blockIdx.x * 8 + wave;        // global tile id
  const int mat = t / (MT * NT);              // 0=Q 1=K 2=V
  const int rt = t - mat * (MT * NT);
  const int mt = rt / NT;                     // token-tile
  const int nt = rt - mt * NT;                // feature-tile

  const float* X = (mat == 0) ? Xq : (mat == 1) ? Xk : Xv;
  const float* W = (mat == 0) ? Wq : (mat == 1) ? Wk : Wv;
  const float* BI = (mat == 0) ? bq : (mat == 1) ? bk : bv;
  float* O = (mat == 0) ? Qh : (mat == 1) ? Kh : Vh;

  const int row0 = mt * 16;
  const int col0 = nt * 16;
  const int n = lane & 15;            // M for A rows, N for B cols
  const int khalf2 = (lane >> 4) * 2; // K sub-offset per lane half

  const float* xp = X + (size_t)(row0 + n) * DM + khalf2;
  const float* wp = W + (size_t)(col0 + n) * DM + khalf2;

  v8f acc = {};
#pragma unroll 8
  for (int k0 = 0; k0 < DM; k0 += 4) {
    v2f a = *(const v2f*)(xp + k0);
    v2f b = *(const v2f*)(wp + k0);
    acc = wmma_f32(a, b, acc);
  }

  const float bias = BI[col0 + n];

  // Scatter C-layout tile into per-head [B,H,S,DK] workspace.
  const int bIdx = row0 / SEQ;                 // tile never straddles batches
  const int s0 = row0 - bIdx * SEQ;
  const int h = col0 / DK;
  const int d = (col0 % DK) + n;
  const int moff = (lane >> 4) * 8;
  float* obase = O + (((size_t)(bIdx * H + h)) * SEQ) * DK + d;
#pragma unroll
  for (int r = 0; r < 8; ++r) {
    obase[(size_t)(s0 + r + moff) * DK] = acc[r] + bias;
  }
}

// ---------------------------------------------------------------------------
// Kernel 2: flash attention per (b,h).  One wave owns 32 query rows (two
// 16-row subtiles sharing every K/V B-operand -> 2x FLOPs per byte of L2
// traffic) and streams all 128 k-tiles with online softmax.  Score tiles
// (C layout) are bounced through per-wave padded LDS buffers to re-enter
// the A layout for P @ V; same-wave LDS ops are in-order on CDNA5, so no
// block barrier is needed.
// ---------------------------------------------------------------------------
__global__ __launch_bounds__(128) void attn_kernel(
    const float* __restrict__ Qh, const float* __restrict__ Kh,
    const float* __restrict__ Vh, const int* __restrict__ mask,
    float* __restrict__ out) {
  __shared__ float pShm[4][2][16][18];  // per-wave, per-subtile, padded rows

  const int lane = threadIdx.x & 31;
  const int wave = threadIdx.x >> 5;
  const int gw = blockIdx.x * 4 + wave;     // 0..2047
  const int qt = gw & 63;                   // 64 q-tiles of 32 rows
  const int bh = gw >> 6;                   // 0..31
  const int b = bh >> 4;
  const int h = bh & 15;
  const int q0 = qt * 32;

  const float* Qp = Qh + ((size_t)bh * SEQ + q0) * DK;
  const float* Kp = Kh + (size_t)bh * SEQ * DK;
  const float* Vp = Vh + (size_t)bh * SEQ * DK;
  const int* Mp = mask + (size_t)b * SEQ * SEQ + (size_t)q0 * SEQ;

  const int n = lane & 15;
  const int half = lane >> 4;
  const int khalf2 = half * 2;
  const int moff = half * 8;

  // Q subtiles preloaded as A-chunks (d split into 16 groups of 4).
  v2f qa[2][16];
#pragma unroll
  for (int sub = 0; sub < 2; ++sub)
#pragma unroll
    for (int j = 0; j < 16; ++j)
      qa[sub][j] =
          *(const v2f*)(Qp + (size_t)(sub * 16 + n) * DK + j * 4 + khalf2);

  v8f oacc[2][4] = {{{}, {}, {}, {}}, {{}, {}, {}, {}}};
  float mrow[2][8], lrow[2][8];
#pragma unroll
  for (int sub = 0; sub < 2; ++sub)
#pragma unroll
    for (int r = 0; r < 8; ++r) {
      mrow[sub][r] = -1.0e30f;
      lrow[sub][r] = 0.0f;
    }

  float(*p)[16][18] = pShm[wave];
  const float scale = 0.125f;  // 1/sqrt(64)

  for (int k0 = 0; k0 < SEQ; k0 += 16) {
    // Speculative prefetch of the next K/V tile rows for this lane.
    __builtin_prefetch(Kp + (size_t)(k0 + 16 + n) * DK + khalf2, 0, 0);
    __builtin_prefetch(Vp + (size_t)(k0 + 16 + n) * DK + khalf2, 0, 0);

    // ---- scores S = Q K^T (two 16x16 tiles, K-dim = 64) ----
    v8f s0 = {}, s1 = {};
    const float* kp = Kp + (size_t)(k0 + n) * DK + khalf2;
#pragma unroll
    for (int j = 0; j < 16; ++j) {
      v2f bK = *(const v2f*)(kp + j * 4);
      s0 = wmma_f32(qa[0][j], bK, s0);
      s1 = wmma_f32(qa[1][j], bK, s1);
    }

    // ---- scale + mask + online softmax per subtile ----
#pragma unroll
    for (int sub = 0; sub < 2; ++sub) {
      v8f& s = sub ? s1 : s0;
#pragma unroll
      for (int r = 0; r < 8; ++r) {
        int mk = Mp[(size_t)(sub * 16 + r + moff) * SEQ + k0 + n];
        s[r] = (mk == 0) ? -1.0e9f : s[r] * scale;
      }
      float alpha[8];
#pragma unroll
      for (int r = 0; r < 8; ++r) {
        float v = s[r];
        v = fmaxf(v, __shfl_xor(v, 1, 32));
        v = fmaxf(v, __shfl_xor(v, 2, 32));
        v = fmaxf(v, __shfl_xor(v, 4, 32));
        v = fmaxf(v, __shfl_xor(v, 8, 32));
        float mnew = fmaxf(mrow[sub][r], v);
        alpha[r] = __expf(mrow[sub][r] - mnew);
        float pv = __expf(s[r] - mnew);
        s[r] = pv;
        mrow[sub][r] = mnew;
        float rs = pv;
        rs += __shfl_xor(rs, 1, 32);
        rs += __shfl_xor(rs, 2, 32);
        rs += __shfl_xor(rs, 4, 32);
        rs += __shfl_xor(rs, 8, 32);
        lrow[sub][r] = lrow[sub][r] * alpha[r] + rs;
      }
      // rescale running output
#pragma unroll
      for (int r = 0; r < 8; ++r) {
        oacc[sub][0][r] *= alpha[r];
        oacc[sub][1][r] *= alpha[r];
        oacc[sub][2][r] *= alpha[r];
        oacc[sub][3][r] *= alpha[r];
      }
      // bounce P through this wave's private LDS tile (C -> A layout);
      // same-wave LDS ops are in-order, no barrier required.
#pragma unroll
      for (int r = 0; r < 8; ++r) p[sub][r + moff][n] = s[r];
    }

    // ---- O += P @ V  (two 16x64 accumulators, K-dim = 16 keys) ----
#pragma unroll
    for (int kk = 0; kk < 16; kk += 4) {
      v2f aP0 = *(const v2f*)(&p[0][n][kk + khalf2]);
      v2f aP1 = *(const v2f*)(&p[1][n][kk + khalf2]);
      const float* vr = Vp + (size_t)(k0 + kk + khalf2) * DK + n;
#pragma unroll
      for (int c = 0; c < 4; ++c) {
        v2f bV;
        bV.x = vr[c * 16];
        bV.y = vr[DK + c * 16];
        oacc[0][c] = wmma_f32(aP0, bV, oacc[0][c]);
        oacc[1][c] = wmma_f32(aP1, bV, oacc[1][c]);
      }
    }
  }

  // ---- normalize and store to [B, S, H*DK] ----
#pragma unroll
  for (int sub = 0; sub < 2; ++sub) {
    float* op =
        out + ((size_t)(b * SEQ + q0 + sub * 16)) * (H * DK) + h * DK + n;
#pragma unroll
    for (int r = 0; r < 8; ++r) {
      float inv = 1.0f / lrow[sub][r];
      size_t rowoff = (size_t)(r + moff) * (H * DK);
#pragma unroll
      for (int c = 0; c < 4; ++c) op[rowoff + c * 16] = oacc[sub][c][r] * inv;
    }
  }
}

extern "C" void kernel_launch(void* const* d_in, const int* in_sizes, int n_in,
                              void* d_out, int out_size, void* d_ws,
                              size_t ws_size, hipStream_t stream) {
  const float* query = (const float*)d_in[0];
  const float* key = (const float*)d_in[1];
  const float* value = (const float*)d_in[2];
  const int* mask = (const int*)d_in[3];
  const float* Wq = (const float*)d_in[4];
  const float* bq = (const float*)d_in[5];
  const float* Wk = (const float*)d_in[6];
  const float* bk = (const float*)d_in[7];
  const float* Wv = (const float*)d_in[8];
  const float* bv = (const float*)d_in[9];
  float* out = (float*)d_out;

  const size_t headElems = (size_t)BATCH * H * SEQ * DK;  // 4M floats
  float* Qh = (float*)d_ws;
  float* Kh = Qh + headElems;
  float* Vh = Kh + headElems;

  // 3 * 256 * 64 tiles, 8 waves (tiles) per 256-thread block
  proj_kernel<<<(3 * MT * NT) / 8, 256, 0, stream>>>(
      query, key, value, Wq, bq, Wk, bk, Wv, bv, Qh, Kh, Vh);

  // 32 (b,h) * 64 q-tiles (32 rows each) = 2048 waves, 4 waves per block
  attn_kernel<<<2048 / 4, 128, 0, stream>>>(Qh, Kh, Vh, mask, out);
}